// MultiHeadAttention_13451837571754
// MI455X (gfx1250) — compile-verified
//
#include <hip/hip_runtime.h>
#include <hip/hip_bf16.h>
#include <math.h>

// ---------------------------------------------------------------------------
// MHA forward (fp32) for MI455X / gfx1250, built on V_WMMA_F32_16X16X4_F32.
// Memory-bound on the 268MB attn tensor -> keep fp32 (faithful) and use the
// fp32 WMMA path for all GEMMs. wave32; one 16x16 output tile per wave.
// ---------------------------------------------------------------------------

#define D_MODEL   512
#define NUM_HEADS 8
#define DEPTH     64
#define BATCH     2
#define SEQ       2048
#define ROWS      (BATCH * SEQ)          // 4096
#define NEG_SCALE (-1e9f)
#define INV_SQRT_D 0.125f                // 1/sqrt(64)

typedef float v2f __attribute__((ext_vector_type(2)));
typedef float v8f __attribute__((ext_vector_type(8)));

// D = A(16x4) * B(4x16) + C   -- fp32 WMMA, wave32, EXEC must be all ones.
__device__ __forceinline__ v8f wmma_step(v2f a, v2f b, v8f c) {
    return __builtin_amdgcn_wmma_f32_16x16x4_f32(
        /*neg_a=*/false, a, /*neg_b=*/false, b,
        /*c_mod=*/(short)0, c, /*reuse_a=*/false, /*reuse_b=*/false);
}

// ---------------------------------------------------------------------------
// Generic [4096x512] @ [512x512] + bias GEMM.
// MODE 0: scatter result to [b, h, s, d]  (head split for Q/K/V projections)
// MODE 1: plain row-major [row, col]      (final Wo projection)
//
// A 16x4 fp32 tile layout (ISA 7.12.2): lanes 0-15 hold M=lane with K=0 (v0),
// K=1 (v1); lanes 16-31 hold K=2,3.  B 4x16 mirrors with lane = N.
// C/D: VGPR r holds M = r + 8*(lane>=16), N = lane&15.
// ---------------------------------------------------------------------------
template <int MODE>
__global__ void gemm512_kernel(const float* __restrict__ X,
                               const float* __restrict__ W,
                               const float* __restrict__ bias,
                               float* __restrict__ Y) {
    const int lane = threadIdx.x & 31;
    const int wave = (blockIdx.x * blockDim.x + threadIdx.x) >> 5;
    const int tilesN = D_MODEL / 16;          // 32
    const int tm = wave / tilesN;             // 0..255
    const int tn = wave % tilesN;             // 0..31
    const int half = lane >> 4;
    const int ml   = lane & 15;

    const float* Xr = X + (size_t)(tm * 16 + ml) * D_MODEL;   // A row for this lane
    const int ncol  = tn * 16 + ml;                           // B column for this lane

    v8f c = {};
#pragma unroll 4
    for (int k0 = 0; k0 < D_MODEL; k0 += 4) {
        const int ka = k0 + 2 * half;
        v2f a, b;
        a.x = Xr[ka];
        a.y = Xr[ka + 1];
        const float* Wk = W + (size_t)ka * D_MODEL + ncol;
        b.x = Wk[0];
        b.y = Wk[D_MODEL];
        c = wmma_step(a, b, c);
    }

    const float bv = bias[ncol];
#pragma unroll
    for (int r = 0; r < 8; ++r) {
        const int m   = r + 8 * half;
        const int row = tm * 16 + m;
        const float val = c[r] + bv;
        if (MODE == 0) {
            const int bb = row >> 11;       // row / SEQ
            const int s  = row & (SEQ - 1);
            const int h  = ncol >> 6;       // col / DEPTH
            const int d  = ncol & (DEPTH - 1);
            Y[(((size_t)(bb * NUM_HEADS + h) * SEQ) + s) * DEPTH + d] = val;
        } else {
            Y[(size_t)row * D_MODEL + ncol] = val;
        }
    }
}

// ---------------------------------------------------------------------------
// logits[bh, k, q] = (KK[bh,k,:] . QQ[bh,q,:]) / 8 + mask[b,0,k,q] * -1e9
// A = KK rows (M = k), B = QQ^T (N = q, inner dim = depth), 16 WMMA steps.
// ---------------------------------------------------------------------------
__global__ void logits_kernel(const float* __restrict__ KK,
                              const float* __restrict__ QQ,
                              const int* __restrict__ mask,
                              float* __restrict__ attn) {
    const int lane = threadIdx.x & 31;
    const int wave = (blockIdx.x * blockDim.x + threadIdx.x) >> 5;
    const int bh = wave >> 14;                 // 16384 tiles per (b,h)
    const int t  = wave & 16383;
    const int tk = t >> 7;                     // 0..127 (k tile)
    const int tq = t & 127;                    // 0..127 (q tile)
    const int half = lane >> 4;
    const int ml   = lane & 15;

    const float* Ak = KK + ((size_t)bh * SEQ + tk * 16 + ml) * DEPTH;
    const float* Bq = QQ + ((size_t)bh * SEQ + tq * 16 + ml) * DEPTH;

    v8f c = {};
#pragma unroll 4
    for (int d0 = 0; d0 < DEPTH; d0 += 4) {
        const int kd = d0 + 2 * half;
        v2f a, b;
        a.x = Ak[kd];
        a.y = Ak[kd + 1];
        b.x = Bq[kd];
        b.y = Bq[kd + 1];
        c = wmma_step(a, b, c);
    }

    const int  b    = bh >> 3;
    const int  qcol = tq * 16 + ml;
    float*       dst = attn + (size_t)bh * SEQ * SEQ;
    const int* mrow  = mask + (size_t)b * SEQ * SEQ;
#pragma unroll
    for (int r = 0; r < 8; ++r) {
        const int krow = tk * 16 + r + 8 * half;
        const float val = c[r] * INV_SQRT_D +
                          (float)mrow[(size_t)krow * SEQ + qcol] * NEG_SCALE;
        dst[(size_t)krow * SEQ + qcol] = val;
    }
}

// ---------------------------------------------------------------------------
// In-place softmax along q (last axis), one 256-thread block per (bh,k) row.
// ---------------------------------------------------------------------------
__global__ void softmax_kernel(float* __restrict__ attn) {
    float* p = attn + (size_t)blockIdx.x * SEQ;
    const int tid = threadIdx.x;
    const int wid = tid >> 5;
    const int ln  = tid & 31;
    __shared__ float sred[8];

    float v[8];
    float m = -INFINITY;
#pragma unroll
    for (int i = 0; i < 8; ++i) {
        v[i] = p[tid + i * 256];
        m = fmaxf(m, v[i]);
    }
    for (int off = 16; off; off >>= 1) m = fmaxf(m, __shfl_xor(m, off, 32));
    if (ln == 0) sred[wid] = m;
    __syncthreads();
    float m0 = sred[0];
#pragma unroll
    for (int w = 1; w < 8; ++w) m0 = fmaxf(m0, sred[w]);
    __syncthreads();

    float s = 0.f;
#pragma unroll
    for (int i = 0; i < 8; ++i) {
        v[i] = __expf(v[i] - m0);
        s += v[i];
    }
    for (int off = 16; off; off >>= 1) s += __shfl_xor(s, off, 32);
    if (ln == 0) sred[wid] = s;
    __syncthreads();
    float s0 = 0.f;
#pragma unroll
    for (int w = 0; w < 8; ++w) s0 += sred[w];
    const float inv = 1.0f / s0;
#pragma unroll
    for (int i = 0; i < 8; ++i) p[tid + i * 256] = v[i] * inv;
}

// ---------------------------------------------------------------------------
// out[bh, q, d] = sum_k attn[bh, k, q] * VV[bh, k, d]  (A^T @ V)
// A' = attn transposed via addressing (M = q), B = VV (N = d), K = 2048.
// Result scattered as concat[b, q, h*64 + d].
// ---------------------------------------------------------------------------
__global__ void av_kernel(const float* __restrict__ attn,
                          const float* __restrict__ VV,
                          float* __restrict__ concat) {
    const int lane = threadIdx.x & 31;
    const int wave = (blockIdx.x * blockDim.x + threadIdx.x) >> 5;
    const int bh = wave >> 9;                  // 512 tiles per (b,h)
    const int t  = wave & 511;
    const int tq = t >> 2;                     // 0..127 (q tile)
    const int td = t & 3;                      // 0..3   (d tile)
    const int half = lane >> 4;
    const int ml   = lane & 15;

    const float* A = attn + (size_t)bh * SEQ * SEQ;        // [k][q]
    const float* B = VV + (size_t)bh * SEQ * DEPTH;        // [k][d]
    const int q    = tq * 16 + ml;                         // A' row for lane
    const int dcol = td * 16 + ml;                         // B col for lane

    v8f c = {};
#pragma unroll 4
    for (int k0 = 0; k0 < SEQ; k0 += 4) {
        const int kk = k0 + 2 * half;
        v2f a, b;
        a.x = A[(size_t)kk * SEQ + q];
        a.y = A[(size_t)(kk + 1) * SEQ + q];
        b.x = B[(size_t)kk * DEPTH + dcol];
        b.y = B[(size_t)(kk + 1) * DEPTH + dcol];
        c = wmma_step(a, b, c);
    }

    const int bb = bh >> 3;
    const int h  = bh & 7;
#pragma unroll
    for (int r = 0; r < 8; ++r) {
        const int qq = tq * 16 + r + 8 * half;
        concat[((size_t)(bb * SEQ + qq)) * D_MODEL + h * DEPTH + dcol] = c[r];
    }
}

// ---------------------------------------------------------------------------
// Launch: inputs in setup_inputs() order:
//   0:v 1:q 2:k 3:mask 4:wq_w 5:wq_b 6:wk_w 7:wk_b 8:wv_w 9:wv_b 10:wo_w 11:wo_b
// d_out = [out (2*2048*512)] ++ [attn (2*8*2048*2048)]  (fp32)
// d_ws: QQ | KK | VV, concat aliases QQ (dead after logits) -> 24MB needed.
// ---------------------------------------------------------------------------
extern "C" void kernel_launch(void* const* d_in, const int* in_sizes, int n_in,
                              void* d_out, int out_size, void* d_ws, size_t ws_size,
                              hipStream_t stream) {
    const float* v    = (const float*)d_in[0];
    const float* q    = (const float*)d_in[1];
    const float* k    = (const float*)d_in[2];
    const int*   mask = (const int*)d_in[3];
    const float* wq_w = (const float*)d_in[4];
    const float* wq_b = (const float*)d_in[5];
    const float* wk_w = (const float*)d_in[6];
    const float* wk_b = (const float*)d_in[7];
    const float* wv_w = (const float*)d_in[8];
    const float* wv_b = (const float*)d_in[9];
    const float* wo_w = (const float*)d_in[10];
    const float* wo_b = (const float*)d_in[11];

    float* out  = (float*)d_out;
    float* attn = out + (size_t)BATCH * SEQ * D_MODEL;      // +2,097,152

    const size_t PROJ = (size_t)BATCH * NUM_HEADS * SEQ * DEPTH;  // 2,097,152
    float* QQ = (float*)d_ws;
    float* KK = QQ + PROJ;
    float* VV = KK + PROJ;
    float* CC = QQ;   // concat aliases QQ: QQ is dead once logits are written

    const dim3 blk(256);   // 8 wave32s per block

    // Projections: 4096x512 / (16x16) = 8192 tiles -> 8192 waves -> 1024 blocks
    gemm512_kernel<0><<<1024, blk, 0, stream>>>(q, wq_w, wq_b, QQ);
    gemm512_kernel<0><<<1024, blk, 0, stream>>>(k, wk_w, wk_b, KK);
    gemm512_kernel<0><<<1024, blk, 0, stream>>>(v, wv_w, wv_b, VV);

    // Logits: 16 bh * 128 * 128 tiles = 262144 waves -> 32768 blocks
    logits_kernel<<<32768, blk, 0, stream>>>(KK, QQ, mask, attn);

    // Softmax along q: one block per (bh,k) row = 32768 rows
    softmax_kernel<<<32768, blk, 0, stream>>>(attn);

    // A^T @ V: 16 bh * 128 * 4 tiles = 8192 waves -> 1024 blocks
    av_kernel<<<1024, blk, 0, stream>>>(attn, VV, CC);

    // Final projection (plain layout) into d_out
    gemm512_kernel<1><<<1024, blk, 0, stream>>>(CC, wo_w, wo_b, out);
}